// DenseGrid_15410342658412
// MI455X (gfx1250) — compile-verified
//
#include <hip/hip_runtime.h>

// ---------------------------------------------------------------------------
// Hybrid grid encoder: trilinear( [8,128,128,128] ) + 3x bilinear( [8,1024,1024] )
// at 2M points -> [N, 8] f32.
//
// Gather-bound (AI ~0.6 FLOP/B). MI455X strategy:
//  * whole table set (64MB grid + 96MB planes) fits the 192MB L2
//  * repack channel-major -> channel-last in d_ws so each corner gather is
//    2x global_load_b128 instead of 8x scattered b32 (160 -> 40 reqs/point)
//  * NT hints on streaming traffic (repack sources, final output) so the
//    packed tables stay L2-resident across graph replays
//  * all gathers use uniform SGPR base + signed 32-bit VGPR byte offset
//    (gfx1250 GVS addressing: 1 VGPR/address, no 64-bit carry chains)
// ---------------------------------------------------------------------------

#define LRES 128
#define HRES 1024
#define CH   8

typedef float v4f __attribute__((ext_vector_type(4)));

__device__ __forceinline__ void to_idx(float c, int size, int& i0, int& i1, float& w) {
    // align_corners=True mapping, matches reference (w from unclipped floor)
    float p = (c + 1.0f) * 0.5f * (float)(size - 1);
    float f = floorf(p);
    w = p - f;
    int ii = (int)f;
    ii = ii < 0 ? 0 : (ii > size - 1 ? size - 1 : ii);
    i0 = ii;
    int j = ii + 1;
    i1 = j > size - 1 ? size - 1 : j;
}

// Load 8 contiguous channels (32B) at a 32-bit byte offset from a uniform base
// and accumulate with weight w into (a0, a1).
__device__ __forceinline__ void gacc8(const float* __restrict__ base, int boff, float w,
                                      v4f& a0, v4f& a1) {
    const v4f* p4 = reinterpret_cast<const v4f*>(reinterpret_cast<const char*>(base) + boff);
    v4f lo = p4[0];
    v4f hi = p4[1];
    a0.x = fmaf(lo.x, w, a0.x); a0.y = fmaf(lo.y, w, a0.y);
    a0.z = fmaf(lo.z, w, a0.z); a0.w = fmaf(lo.w, w, a0.w);
    a1.x = fmaf(hi.x, w, a1.x); a1.y = fmaf(hi.y, w, a1.y);
    a1.z = fmaf(hi.z, w, a1.z); a1.w = fmaf(hi.w, w, a1.w);
}

// -------------------- repack kernels (channel-major -> channel-last) --------

__global__ __launch_bounds__(256)
void dg_repack_grid(const float* __restrict__ g, float* __restrict__ o) {
    // g: [8][128^3]  ->  o: [128^3][8]
    unsigned v = blockIdx.x * blockDim.x + threadIdx.x;
    const unsigned NV = (unsigned)LRES * LRES * LRES;   // 2,097,152
    if (v >= NV) return;
    v4f lo, hi;
    lo.x = __builtin_nontemporal_load(g + (size_t)0 * NV + v);
    lo.y = __builtin_nontemporal_load(g + (size_t)1 * NV + v);
    lo.z = __builtin_nontemporal_load(g + (size_t)2 * NV + v);
    lo.w = __builtin_nontemporal_load(g + (size_t)3 * NV + v);
    hi.x = __builtin_nontemporal_load(g + (size_t)4 * NV + v);
    hi.y = __builtin_nontemporal_load(g + (size_t)5 * NV + v);
    hi.z = __builtin_nontemporal_load(g + (size_t)6 * NV + v);
    hi.w = __builtin_nontemporal_load(g + (size_t)7 * NV + v);
    v4f* p = reinterpret_cast<v4f*>(o + (size_t)v * CH);
    p[0] = lo;
    p[1] = hi;
}

__global__ __launch_bounds__(256)
void dg_repack_planes(const float* __restrict__ g, float* __restrict__ o) {
    // g: [3][8][1024^2]  ->  o: [3][1024^2][8]   (blockIdx.y = plane)
    unsigned pix = blockIdx.x * blockDim.x + threadIdx.x;
    const unsigned NP = (unsigned)HRES * HRES;          // 1,048,576
    if (pix >= NP) return;
    unsigned pl = blockIdx.y;
    const float* src = g + (size_t)pl * CH * NP + pix;
    v4f lo, hi;
    lo.x = __builtin_nontemporal_load(src + (size_t)0 * NP);
    lo.y = __builtin_nontemporal_load(src + (size_t)1 * NP);
    lo.z = __builtin_nontemporal_load(src + (size_t)2 * NP);
    lo.w = __builtin_nontemporal_load(src + (size_t)3 * NP);
    hi.x = __builtin_nontemporal_load(src + (size_t)4 * NP);
    hi.y = __builtin_nontemporal_load(src + (size_t)5 * NP);
    hi.z = __builtin_nontemporal_load(src + (size_t)6 * NP);
    hi.w = __builtin_nontemporal_load(src + (size_t)7 * NP);
    v4f* p = reinterpret_cast<v4f*>(o + ((size_t)pl * NP + pix) * CH);
    p[0] = lo;
    p[1] = hi;
}

// -------------------- main sampler (packed, channel-last) -------------------

__global__ __launch_bounds__(256)
void dg_sample_packed(const float* __restrict__ xyz, const int* __restrict__ boundp,
                      const float* __restrict__ Lp, const float* __restrict__ Pp,
                      float* __restrict__ out, int n) {
    int i = blockIdx.x * blockDim.x + threadIdx.x;
    if (i >= n) return;

    float inv = 1.0f / (float)boundp[0];
    const float* cp = xyz + 3 * (size_t)i;
    float x = cp[0] * inv;
    float y = cp[1] * inv;
    float z = cp[2] * inv;

    // Two accumulator pairs (grid / planes) to halve the serial FMA chain.
    v4f t0 = {0.f, 0.f, 0.f, 0.f}, t1 = {0.f, 0.f, 0.f, 0.f};
    v4f q0 = {0.f, 0.f, 0.f, 0.f}, q1 = {0.f, 0.f, 0.f, 0.f};

    // ---- trilinear from 128^3 grid ----
    int x0, x1, y0, y1, z0, z1;
    float wx, wy, wz;
    to_idx(x, LRES, x0, x1, wx);
    to_idx(y, LRES, y0, y1, wy);
    to_idx(z, LRES, z0, z1, wz);
    float ux = 1.0f - wx, uy = 1.0f - wy, uz = 1.0f - wz;

    const int VB = CH * (int)sizeof(float);             // 32 bytes per voxel
    auto vb = [VB](int zi, int yi, int xi) {
        return (((zi * LRES) + yi) * LRES + xi) * VB;   // max 64MB-32, fits i32
    };
    gacc8(Lp, vb(z0, y0, x0), uz * uy * ux, t0, t1);
    gacc8(Lp, vb(z0, y0, x1), uz * uy * wx, t0, t1);
    gacc8(Lp, vb(z0, y1, x0), uz * wy * ux, t0, t1);
    gacc8(Lp, vb(z0, y1, x1), uz * wy * wx, t0, t1);
    gacc8(Lp, vb(z1, y0, x0), wz * uy * ux, t0, t1);
    gacc8(Lp, vb(z1, y0, x1), wz * uy * wx, t0, t1);
    gacc8(Lp, vb(z1, y1, x0), wz * wy * ux, t0, t1);
    gacc8(Lp, vb(z1, y1, x1), wz * wy * wx, t0, t1);

    // ---- three bilinear planes at 1024^2 ----
    int xh0, xh1, yh0, yh1, zh0, zh1;
    float wxh, wyh, wzh;
    to_idx(x, HRES, xh0, xh1, wxh);
    to_idx(y, HRES, yh0, yh1, wyh);
    to_idx(z, HRES, zh0, zh1, wzh);

    auto bil = [&](const float* __restrict__ base, int w0, int w1, float ww,
                   int h0, int h1, float wh) {
        float uw = 1.0f - ww, uh = 1.0f - wh;
        gacc8(base, (h0 * HRES + w0) * VB, uh * uw, q0, q1);
        gacc8(base, (h0 * HRES + w1) * VB, uh * ww, q0, q1);
        gacc8(base, (h1 * HRES + w0) * VB, wh * uw, q0, q1);
        gacc8(base, (h1 * HRES + w1) * VB, wh * ww, q0, q1);
    };
    const size_t PSZ = (size_t)HRES * HRES * CH;
    const float* P0 = Pp;                               // plane 0: (x->W, y->H)
    const float* P1 = Pp + PSZ;                         // plane 1: (y->W, z->H)
    const float* P2 = Pp + 2 * PSZ;                     // plane 2: (z->W, x->H)
    bil(P0, xh0, xh1, wxh, yh0, yh1, wyh);
    bil(P1, yh0, yh1, wyh, zh0, zh1, wzh);
    bil(P2, zh0, zh1, wzh, xh0, xh1, wxh);

    v4f a0 = {t0.x + q0.x, t0.y + q0.y, t0.z + q0.z, t0.w + q0.w};
    v4f a1 = {t1.x + q1.x, t1.y + q1.y, t1.z + q1.z, t1.w + q1.w};

    // ---- streaming NT store of [N,8] output ----
    v4f* o = reinterpret_cast<v4f*>(out + (size_t)i * CH);
    __builtin_nontemporal_store(a0, o);
    __builtin_nontemporal_store(a1, o + 1);
}

// -------------------- fallback: sample directly from original layout -------

__global__ __launch_bounds__(256)
void dg_sample_direct(const float* __restrict__ xyz, const int* __restrict__ boundp,
                      const float* __restrict__ L, const float* __restrict__ P,
                      float* __restrict__ out, int n) {
    int i = blockIdx.x * blockDim.x + threadIdx.x;
    if (i >= n) return;

    float inv = 1.0f / (float)boundp[0];
    const float* cp = xyz + 3 * (size_t)i;
    float x = cp[0] * inv;
    float y = cp[1] * inv;
    float z = cp[2] * inv;

    float acc[CH];
#pragma unroll
    for (int c = 0; c < CH; ++c) acc[c] = 0.0f;

    int x0, x1, y0, y1, z0, z1;
    float wx, wy, wz;
    to_idx(x, LRES, x0, x1, wx);
    to_idx(y, LRES, y0, y1, wy);
    to_idx(z, LRES, z0, z1, wz);
    float ux = 1.0f - wx, uy = 1.0f - wy, uz = 1.0f - wz;

    const size_t SL = (size_t)LRES * LRES * LRES;
    auto gatherL = [&](int zi, int yi, int xi, float w) {
        size_t off = (size_t)((zi * LRES + yi) * LRES + xi);
#pragma unroll
        for (int c = 0; c < CH; ++c) acc[c] = fmaf(L[(size_t)c * SL + off], w, acc[c]);
    };
    gatherL(z0, y0, x0, uz * uy * ux);
    gatherL(z0, y0, x1, uz * uy * wx);
    gatherL(z0, y1, x0, uz * wy * ux);
    gatherL(z0, y1, x1, uz * wy * wx);
    gatherL(z1, y0, x0, wz * uy * ux);
    gatherL(z1, y0, x1, wz * uy * wx);
    gatherL(z1, y1, x0, wz * wy * ux);
    gatherL(z1, y1, x1, wz * wy * wx);

    int xh0, xh1, yh0, yh1, zh0, zh1;
    float wxh, wyh, wzh;
    to_idx(x, HRES, xh0, xh1, wxh);
    to_idx(y, HRES, yh0, yh1, wyh);
    to_idx(z, HRES, zh0, zh1, wzh);

    const size_t SP = (size_t)HRES * HRES;
    auto gatherP = [&](const float* base, int hi, int wi, float w) {
        size_t off = (size_t)hi * HRES + wi;
#pragma unroll
        for (int c = 0; c < CH; ++c) acc[c] = fmaf(base[(size_t)c * SP + off], w, acc[c]);
    };
    auto bil = [&](const float* base, int w0, int w1, float ww, int h0, int h1, float wh) {
        float uw = 1.0f - ww, uh = 1.0f - wh;
        gatherP(base, h0, w0, uh * uw);
        gatherP(base, h0, w1, uh * ww);
        gatherP(base, h1, w0, wh * uw);
        gatherP(base, h1, w1, wh * ww);
    };
    bil(P + (size_t)0 * CH * SP, xh0, xh1, wxh, yh0, yh1, wyh);
    bil(P + (size_t)1 * CH * SP, yh0, yh1, wyh, zh0, zh1, wzh);
    bil(P + (size_t)2 * CH * SP, zh0, zh1, wzh, xh0, xh1, wxh);

#pragma unroll
    for (int c = 0; c < CH; ++c)
        __builtin_nontemporal_store(acc[c], out + (size_t)i * CH + c);
}

// ---------------------------------------------------------------------------

extern "C" void kernel_launch(void* const* d_in, const int* in_sizes, int n_in,
                              void* d_out, int out_size, void* d_ws, size_t ws_size,
                              hipStream_t stream) {
    const float* xyz   = (const float*)d_in[0];
    const int*   bound = (const int*)d_in[1];
    const float* L     = (const float*)d_in[2];
    const float* P     = (const float*)d_in[3];
    float*       out   = (float*)d_out;

    int n = in_sizes[0] / 3;

    const size_t LVOX   = (size_t)LRES * LRES * LRES;       // 2,097,152 voxels
    const size_t LBYTES = LVOX * CH * sizeof(float);        // 64 MB
    const size_t PPIX   = (size_t)HRES * HRES;              // 1,048,576 pixels
    const size_t PBYTES = 3 * PPIX * CH * sizeof(float);    // 96 MB

    if (ws_size >= LBYTES + PBYTES) {
        float* Lp = (float*)d_ws;
        float* Pp = (float*)((char*)d_ws + LBYTES);
        dg_repack_grid<<<dim3((unsigned)((LVOX + 255) / 256)), 256, 0, stream>>>(L, Lp);
        dg_repack_planes<<<dim3((unsigned)((PPIX + 255) / 256), 3), 256, 0, stream>>>(P, Pp);
        dg_sample_packed<<<(n + 255) / 256, 256, 0, stream>>>(xyz, bound, Lp, Pp, out, n);
    } else {
        dg_sample_direct<<<(n + 255) / 256, 256, 0, stream>>>(xyz, bound, L, P, out, n);
    }
}